// idct_54271206752953
// MI455X (gfx1250) — compile-verified
//
#include <hip/hip_runtime.h>
#include <math.h>

// CDNA5 (gfx1250) wave32 WMMA f32 path.
typedef __attribute__((ext_vector_type(2))) float v2f;
typedef __attribute__((ext_vector_type(8))) float v8f;

#define LSTRIDE 20  // LDS row stride (floats): conflict-free b64 reads, 16B-aligned rows

__global__ __launch_bounds__(256) void idct8x8_wmma_kernel(
    const float* __restrict__ images,
    const float* __restrict__ alpha,
    const float* __restrict__ basis,
    float* __restrict__ out,
    int numGroups)  // one group = 4 consecutive 8x8 tiles = 256 floats
{
  __shared__ float lds[8 * 16 * LSTRIDE];  // per-wave 16x20 scratch, 8 waves/block

  const int lane = threadIdx.x & 31;
  const int wave = threadIdx.x >> 5;
  const int np   = lane & 15;   // N column (B/C/D) and M row (A)
  const int hi   = lane >> 4;   // lane half selects K pairs / M+8

  // Constant A operand: Gbt = blockdiag(G', G'), G'[u][x] = 0.5*a[x]*cx[x][u].
  // a[x] = sqrt(alpha[x*8+x]); cx[x][u] = basis[x,0,u,0] = basis[x*512 + u*8].
  // A-layout (16x4 f32): lane m = lane&15; K = vgpr + 2*hi  (chunk k covers K=4k..4k+3)
  v2f A[4];
#pragma unroll
  for (int k = 0; k < 4; ++k) {
#pragma unroll
    for (int v = 0; v < 2; ++v) {
      const int c = 4 * k + 2 * hi + v;  // column of Gbt
      const int m = np;                  // row of Gbt
      float g = 0.0f;
      if ((m < 8) == (c < 8)) {          // block-diagonal
        const int u = m & 7, x = c & 7;
        g = 0.5f * sqrtf(alpha[x * 8 + x]) * basis[x * 512 + u * 8];
      }
      A[k][v] = g;
    }
  }

  float* wbuf = lds + wave * 16 * LSTRIDE;
  const int nWaves = gridDim.x * 8;
  const int wgid   = blockIdx.x * 8 + wave;

  for (int g = wgid; g < numGroups; g += nWaves) {  // wave-uniform: EXEC stays full
    const float* src = images + (size_t)g * 256;

    // B operand #1: P^T chunks.  B[kk][n] = P[n][4k+2hi+v]; the two v's are
    // adjacent floats in memory -> one global_load_b64 per chunk.
    // P[r][c] lives at tile t=2*(r/8)+(c/8), element (r%8, c%8).
    v2f B1[4];
#pragma unroll
    for (int k = 0; k < 4; ++k) {
      const int c   = 4 * k + 2 * hi;
      const int off = (2 * (np >> 3) + (c >> 3)) * 64 + (np & 7) * 8 + (c & 7);
      B1[k] = *(const v2f*)(src + off);
    }

    // W = Gbt * P^T  (16x16x16 via 4x v_wmma_f32_16x16x4_f32)
    v8f acc = {0.f, 0.f, 0.f, 0.f, 0.f, 0.f, 0.f, 0.f};
#pragma unroll
    for (int k = 0; k < 4; ++k)
      acc = __builtin_amdgcn_wmma_f32_16x16x4_f32(
          false, A[k], false, B1[k], (short)0, acc, false, false);

    // Mid transpose through per-wave LDS: store W row-major (D layout:
    // lane holds W[v+8hi][np]).  DS ops are in-order per wave -> no barrier.
    #pragma unroll
    for (int v = 0; v < 8; ++v)
      wbuf[(v + 8 * hi) * LSTRIDE + np] = acc[v];

    // B operand #2: W^T chunks.  B[kk][n] = W[n][4k+2hi+v] -> adjacent floats
    // in LDS row np -> one ds_load_b64 per chunk (bank-conflict-free, stride 20).
    v2f B2[4];
#pragma unroll
    for (int k = 0; k < 4; ++k) {
      const int c = 4 * k + 2 * hi;
      B2[k] = *(const v2f*)(wbuf + np * LSTRIDE + c);
    }

    // Out = Gbt * W^T + 128   (bias folded into the accumulator seed)
    v8f res = {128.f, 128.f, 128.f, 128.f, 128.f, 128.f, 128.f, 128.f};
#pragma unroll
    for (int k = 0; k < 4; ++k)
      res = __builtin_amdgcn_wmma_f32_16x16x4_f32(
          false, A[k], false, B2[k], (short)0, res, false, false);

    // Store from D layout: Out[v+8hi][np] -> tile t=2hi+np/8, elem (v, np%8).
    float* dst = out + (size_t)g * 256;
    const int t = 2 * hi + (np >> 3);
#pragma unroll
    for (int v = 0; v < 8; ++v)
      dst[t * 64 + v * 8 + (np & 7)] = res[v];
  }
}

extern "C" void kernel_launch(void* const* d_in, const int* in_sizes, int n_in,
                              void* d_out, int out_size, void* d_ws, size_t ws_size,
                              hipStream_t stream) {
  const float* images = (const float*)d_in[0];  // [128,4096,8,8] f32
  const float* alpha  = (const float*)d_in[1];  // [8,8] f32
  const float* basis  = (const float*)d_in[2];  // [8,8,8,8] f32
  float* outp = (float*)d_out;

  const int numGroups = in_sizes[0] / 256;      // 4 tiles (256 floats) per group
  int blocks = 2048;                            // 16384 waves, grid-stride over groups
  const int maxBlocks = (numGroups + 7) / 8;
  if (blocks > maxBlocks) blocks = maxBlocks;
  if (blocks < 1) blocks = 1;

  idct8x8_wmma_kernel<<<blocks, 256, 0, stream>>>(images, alpha, basis, outp, numGroups);
}